// BatchMultiHeadGraphAttention_5738076307606
// MI455X (gfx1250) — compile-verified
//
#include <hip/hip_runtime.h>
#include <hip/hip_bf16.h>

typedef __attribute__((ext_vector_type(2))) float v2f;
typedef __attribute__((ext_vector_type(8))) float v8f;

constexpr int B    = 4;
constexpr int N    = 2048;
constexpr int H    = 4;
constexpr int FIN  = 64;
constexpr int FOUT = 64;
constexpr float NEG_SLOPE = 0.2f;
constexpr float NEG_BIG   = -1e30f;   // finite "-inf": keeps exp() NaN-free

// ---------------------------------------------------------------------------
// Stage 1: h_prime[b,h] = h[b] @ w[h]   (WMMA f32 16x16x4, K=64)
//          attn_src[b,h,n] = tanh(h_prime) . a_src[h]
//          attn_dst[b,h,n] = tanh(h_prime) . a_dst[h]
// One wave32 per (b,h, 16-row tile).  2048 waves total.
// ---------------------------------------------------------------------------
__global__ void gat_stage1(const float* __restrict__ h,
                           const float* __restrict__ w,
                           const float* __restrict__ a_src,
                           const float* __restrict__ a_dst,
                           float* __restrict__ hp,     // [B,H,N,FOUT]
                           float* __restrict__ srcv,   // [B,H,N]
                           float* __restrict__ dstv)   // [B,H,N]
{
    const int lane  = threadIdx.x & 31;
    const int wave  = threadIdx.x >> 5;
    const int wgl   = blockIdx.x * (blockDim.x >> 5) + wave;

    const int rowTile = wgl & (N / 16 - 1);     // 128 tiles
    const int bh      = wgl >> 7;               // 0..15
    const int hh      = bh & (H - 1);
    const int b       = bh >> 2;
    const int rows    = rowTile * 16;

    const int mlo = lane & 15;                  // M (A/C row group) / N col
    const int hi  = lane >> 4;                  // half-wave
    const int ko  = hi * 2;                     // K sub-offset in A/B frags

    const float* hRow = h + ((size_t)b * N + rows + mlo) * FIN;
    const float* wB   = w + (size_t)hh * FIN * FOUT;

    v8f zero = {0.f,0.f,0.f,0.f,0.f,0.f,0.f,0.f};
    v8f acc[4] = {zero, zero, zero, zero};

    // GEMM: 16 k-steps of V_WMMA_F32_16X16X4_F32 per column tile
    for (int k = 0; k < FIN; k += 4) {
        v2f af;
        af.x = hRow[k + ko];
        af.y = hRow[k + ko + 1];
        const float* w0 = wB + (size_t)(k + ko) * FOUT;
#pragma unroll
        for (int c = 0; c < 4; ++c) {
            v2f bf;
            bf.x = w0[c * 16 + mlo];
            bf.y = w0[FOUT + c * 16 + mlo];
            acc[c] = __builtin_amdgcn_wmma_f32_16x16x4_f32(
                false, af, false, bf, (short)0, acc[c], false, false);
        }
    }

    // store h_prime tile + tanh reductions against a_src/a_dst
    const float* as = a_src + hh * FOUT;
    const float* ad = a_dst + hh * FOUT;
    float sp[8], dp[8];
#pragma unroll
    for (int v = 0; v < 8; ++v) { sp[v] = 0.f; dp[v] = 0.f; }

    float* hpB = hp + (size_t)bh * N * FOUT;
#pragma unroll
    for (int c = 0; c < 4; ++c) {
        const int n = c * 16 + mlo;
        const float asn = as[n], adn = ad[n];
#pragma unroll
        for (int v = 0; v < 8; ++v) {
            const int r = rows + v + 8 * hi;        // C-layout row
            const float val = acc[c][v];
            hpB[(size_t)r * FOUT + n] = val;
            const float t = tanhf(val);
            sp[v] += t * asn;
            dp[v] += t * adn;
        }
    }
    // reduce over the 16 lanes of each half-wave (rows v / v+8)
#pragma unroll
    for (int v = 0; v < 8; ++v) {
#pragma unroll
        for (int d = 1; d < 16; d <<= 1) {
            sp[v] += __shfl_xor(sp[v], d, 32);
            dp[v] += __shfl_xor(dp[v], d, 32);
        }
    }
    if (mlo == 0) {      // lanes 0 and 16 hold full sums for rows v / v+8
        float* s0 = srcv + (size_t)bh * N + rows + 8 * hi;
        float* d0 = dstv + (size_t)bh * N + rows + 8 * hi;
#pragma unroll
        for (int v = 0; v < 8; ++v) { s0[v] = sp[v]; d0[v] = dp[v]; }
    }
}

// ---------------------------------------------------------------------------
// Stage 2: flash-style masked softmax + (attn @ h_prime) via WMMA f32.
// 8 waves / block, all sharing one (b,h): the 16x64 V tile is DMA'd to LDS
// once per block with gfx1250 async global->LDS loads (ASYNCcnt), double-
// buffered so the next tile's DMA overlaps the current tile's WMMA work.
// ---------------------------------------------------------------------------
__device__ __forceinline__ void async_tile_to_lds(const float* __restrict__ gbase,
                                                  float* lds_dst)
{
    // 256 threads x 16B = one 16x64 f32 tile, row-major
    const int t  = threadIdx.x;          // 0..255
    const int r  = t >> 4;               // tile row 0..15
    const int c4 = (t & 15) << 2;        // col 0,4,..,60
    const uint64_t ga = (uint64_t)(gbase + (size_t)r * FOUT + c4);
    // LDS byte address = low 32 bits of the generic (flat) shared pointer
    const unsigned la = (unsigned)(uintptr_t)(lds_dst + r * FOUT + c4);
    asm volatile("global_load_async_to_lds_b128 %0, %1, off"
                 :: "v"(la), "v"(ga) : "memory");
}

__device__ __forceinline__ void wait_async_done()
{
    asm volatile("s_wait_asynccnt 0x0" ::: "memory");
}

__global__ void gat_stage2(const float* __restrict__ adj,
                           const float* __restrict__ hp,
                           const float* __restrict__ srcv,
                           const float* __restrict__ dstv,
                           const float* __restrict__ bias,
                           float* __restrict__ out)
{
    __shared__ float ldsV[2][16 * FOUT];          // double-buffered V tile (8KB)

    const int lane = threadIdx.x & 31;
    const int wave = threadIdx.x >> 5;
    const int wgl  = blockIdx.x * (blockDim.x >> 5) + wave;

    const int rowTile = wgl & (N / 16 - 1);
    const int bh      = wgl >> 7;
    const int b       = bh >> 2;
    const int rows    = rowTile * 16;

    const int mlo = lane & 15;
    const int hi  = lane >> 4;
    const int ko  = hi * 2;

    const float* adjRow = adj + ((size_t)b * N + rows + mlo) * N;
    const float  srcm   = srcv[(size_t)bh * N + rows + mlo];
    const float* dstB   = dstv + (size_t)bh * N;
    const float* hpB    = hp + (size_t)bh * N * FOUT;

    v8f zero = {0.f,0.f,0.f,0.f,0.f,0.f,0.f,0.f};
    v8f acc[4] = {zero, zero, zero, zero};
    float rmax = NEG_BIG;    // running row max (row = mlo, mirrored in both halves)
    float rsum = 0.f;        // running row sum

    // prologue: DMA tile 0 into buffer 0
    async_tile_to_lds(hpB, &ldsV[0][0]);

    for (int jt = 0; jt < N; jt += 16) {
        const int ibuf = (jt >> 4) & 1;

        // publish tile jt: own async loads done, then block-wide barrier.
        // The same barrier also fences reuse of the other buffer, so the
        // next DMA below cannot race readers of the previous iteration.
        wait_async_done();
        __syncthreads();
        if (jt + 16 < N)
            async_tile_to_lds(hpB + (size_t)(jt + 16) * FOUT, &ldsV[ibuf ^ 1][0]);

        if (jt + 16 < N)     // wave-uniform; speculative prefetch of next adj tile
            __builtin_prefetch(adjRow + jt + 16, 0, 3);

        // 8 score slots per lane == the A-fragment slots of 4 WMMA k-steps
        float s8[8];
        float tmax = NEG_BIG;
#pragma unroll
        for (int kk = 0; kk < 4; ++kk) {
#pragma unroll
            for (int e = 0; e < 2; ++e) {
                const int jl = kk * 4 + ko + e;
                const float a  = adjRow[jt + jl];
                float sc = srcm + dstB[jt + jl];
                sc = (sc >= 0.f) ? sc : NEG_SLOPE * sc;      // leaky relu
                sc = (a > 0.f) ? sc : NEG_BIG;               // adjacency mask
                s8[kk * 2 + e] = sc;
                tmax = fmaxf(tmax, sc);
            }
        }
        // full row max: pair the two half-waves covering the same row
        tmax = fmaxf(tmax, __shfl_xor(tmax, 16, 32));
        const float newmax = fmaxf(rmax, tmax);
        const float corr   = __expf(rmax - newmax);          // finite args -> no NaN
        rmax = newmax;

        float psum = 0.f;
        v2f  pf[4];
#pragma unroll
        for (int kk = 0; kk < 4; ++kk) {
            const float s0 = s8[kk * 2 + 0];
            const float s1 = s8[kk * 2 + 1];
            const float p0 = (s0 <= NEG_BIG) ? 0.f : __expf(s0 - newmax);
            const float p1 = (s1 <= NEG_BIG) ? 0.f : __expf(s1 - newmax);
            pf[kk].x = p0;
            pf[kk].y = p1;
            psum += p0 + p1;
        }
        psum += __shfl_xor(psum, 16, 32);
        rsum = rsum * corr + psum;

        // rescale accumulator rows: corr for row r lives in lanes r and r+16
#pragma unroll
        for (int v = 0; v < 8; ++v) {
            const float cv = __shfl(corr, v + 8 * hi, 32);
#pragma unroll
            for (int c = 0; c < 4; ++c) acc[c][v] *= cv;
        }

        // acc += P(16x16) @ V(16x64): 4 k-steps x 4 column tiles of wmma f32.
        // V fragments come from LDS (ds_load pairs 256B apart).
        const float* vb = &ldsV[ibuf][0];
#pragma unroll
        for (int kk = 0; kk < 4; ++kk) {
            const int kr = (kk * 4 + ko) * FOUT;
#pragma unroll
            for (int c = 0; c < 4; ++c) {
                const int n = c * 16 + mlo;
                v2f bf;
                bf.x = vb[kr + n];
                bf.y = vb[kr + FOUT + n];
                acc[c] = __builtin_amdgcn_wmma_f32_16x16x4_f32(
                    false, pf[kk], false, bf, (short)0, acc[c], false, false);
            }
        }
    }

    // finalize: divide by row sum, add bias, store [B,H,N,FOUT]
    float* outB = out + (size_t)bh * N * FOUT;
#pragma unroll
    for (int v = 0; v < 8; ++v) {
        const float lv  = __shfl(rsum, v + 8 * hi, 32);
        const float inv = 1.f / lv;
        const int   r   = rows + v + 8 * hi;
#pragma unroll
        for (int c = 0; c < 4; ++c) {
            const int n = c * 16 + mlo;
            outB[(size_t)r * FOUT + n] = acc[c][v] * inv + bias[n];
        }
    }
}

// ---------------------------------------------------------------------------
extern "C" void kernel_launch(void* const* d_in, const int* in_sizes, int n_in,
                              void* d_out, int out_size, void* d_ws, size_t ws_size,
                              hipStream_t stream) {
    const float* h     = (const float*)d_in[0];
    const float* adj   = (const float*)d_in[1];
    const float* w     = (const float*)d_in[2];
    const float* a_src = (const float*)d_in[3];
    const float* a_dst = (const float*)d_in[4];
    const float* bias  = (const float*)d_in[5];
    float* out = (float*)d_out;

    float* ws   = (float*)d_ws;
    float* hp   = ws;                                   // B*H*N*FOUT floats
    float* srcv = hp + (size_t)B * H * N * FOUT;        // B*H*N
    float* dstv = srcv + (size_t)B * H * N;             // B*H*N

    const int wavesTotal    = B * H * (N / 16);         // 2048
    const int wavesPerBlock = 8;                        // 256 threads
    dim3 block(32 * wavesPerBlock);
    dim3 grid(wavesTotal / wavesPerBlock);

    gat_stage1<<<grid, block, 0, stream>>>(h, w, a_src, a_dst, hp, srcv, dstv);
    gat_stage2<<<grid, block, 0, stream>>>(adj, hp, srcv, dstv, bias, out);
}